// GAT_46076409151706
// MI455X (gfx1250) — compile-verified
//
#include <hip/hip_runtime.h>
#include <hip/hip_bf16.h>
#include <math.h>

// ---------------------------------------------------------------------------
// GAT (2-layer, PyG GATConv semantics) for gfx1250 / MI455X.
// Dense projections via V_WMMA_F32_16X16X4_F32 (fp32-exact WMMA path),
// 2 row-tiles per wave for B-fragment reuse, LDS-staged A with bank padding.
// Edge softmax + scatter via L2-resident float atomics (N*128 fits in 192MB L2).
// ---------------------------------------------------------------------------

typedef __attribute__((ext_vector_type(2))) float v2f;
typedef __attribute__((ext_vector_type(8))) float v8f;

#define NEG_SLOPE_ATT 0.2f
#define NEG_SLOPE_ACT 0.01f

// Monotonic float <-> ordered-u32 mapping for atomicMax on floats.
__device__ __forceinline__ unsigned f2ord(float f) {
    unsigned u = __float_as_uint(f);
    return (u & 0x80000000u) ? ~u : (u | 0x80000000u);
}
__device__ __forceinline__ float ord2f(unsigned o) {
    return __uint_as_float((o & 0x80000000u) ? (o & 0x7fffffffu) : ~o);
}
#define ORD_NEG_INF 0x007FFFFFu   // f2ord(-inf)

// ---------------------------------------------------------------------------
// Generic fill (grid-stride).
// ---------------------------------------------------------------------------
__global__ void gat_fill_u32(unsigned* __restrict__ p, unsigned v, size_t n) {
    size_t i = (size_t)blockIdx.x * blockDim.x + threadIdx.x;
    size_t stride = (size_t)gridDim.x * blockDim.x;
    for (; i < n; i += stride) p[i] = v;
}

// ---------------------------------------------------------------------------
// WMMA fp32 GEMM: C[Nrows x Ncols] = A[Nrows x K] * B[K x Ncols].
// One block per 32-row super-tile; wave w owns 16-column tile w and computes
// TWO 16x16 output tiles (rows tileM..+15 and tileM+16..+31) so every B
// fragment load feeds two v_wmma_f32_16x16x4_f32 issues.
// A staged in LDS with +4 row padding (banks 4*r apart -> conflict-free).
//
// VGPR layouts per CDNA5 ISA 7.12.2 (wave32):
//   A 16x4 f32 : lanes 0-15 row=lane {K=k0,k0+1}; lanes 16-31 {K=k0+2,k0+3}
//   B 4x16 f32 : VGPR0 = row (k0 + 2*laneHalf), VGPR1 = row +1, col = lane&15
//   C 16x16    : VGPR r -> M = r + 8*laneHalf, N = lane&15
// ---------------------------------------------------------------------------
__global__ __launch_bounds__(256)
void gat_gemm_wmma(const float* __restrict__ A, const float* __restrict__ B,
                   float* __restrict__ C, int Nrows, int K, int Ncols) {
    extern __shared__ float ldsA[];            // 32 * (K+4) floats
    const int KP    = K + 4;                   // padded LDS row stride
    const int tileM = blockIdx.x * 32;
    const int lane  = threadIdx.x & 31;
    const int wave  = threadIdx.x >> 5;

    // Cooperative stage of the 32 x K A-tile into LDS (padded rows).
    for (int idx = threadIdx.x; idx < 32 * K; idx += blockDim.x) {
        int r = idx / K, c = idx - r * K;
        int row = tileM + r;
        ldsA[r * KP + c] = (row < Nrows) ? A[(size_t)row * K + c] : 0.0f;
    }
    __syncthreads();

    const int tileN = wave * 16;
    if (tileN >= Ncols) return;

    const int rsel  = lane & 15;               // row (A) / col (B,C) select
    const int khalf = (lane >> 4) << 1;        // 0 or 2

    v8f acc0 = {};                             // rows tileM .. tileM+15
    v8f acc1 = {};                             // rows tileM+16 .. tileM+31
    const float* __restrict__ a0p = &ldsA[rsel * KP + khalf];
    const float* __restrict__ a1p = &ldsA[(rsel + 16) * KP + khalf];
    for (int k0 = 0; k0 < K; k0 += 4) {
        v2f a0, a1, b;
        a0.x = a0p[k0];     a0.y = a0p[k0 + 1];
        a1.x = a1p[k0];     a1.y = a1p[k0 + 1];
        b.x  = B[(size_t)(k0 + khalf) * Ncols + tileN + rsel];
        b.y  = B[(size_t)(k0 + khalf + 1) * Ncols + tileN + rsel];
        acc0 = __builtin_amdgcn_wmma_f32_16x16x4_f32(
            false, a0, false, b, (short)0, acc0, false, false);
        acc1 = __builtin_amdgcn_wmma_f32_16x16x4_f32(
            false, a1, false, b, (short)0, acc1, false, false);
    }

    const int mbase = tileM + ((lane >> 4) << 3);
    const int col   = tileN + rsel;
    for (int r = 0; r < 8; ++r) {
        int row0 = mbase + r;
        int row1 = row0 + 16;
        if (row0 < Nrows) C[(size_t)row0 * Ncols + col] = acc0[r];
        if (row1 < Nrows) C[(size_t)row1 * Ncols + col] = acc1[r];
    }
}

// ---------------------------------------------------------------------------
// Attention logits: e_src[n,h] = <h[n,h,:], a_src[h,:]>, same for e_dst.
// One thread per (node, head); C is a multiple of 4.
// ---------------------------------------------------------------------------
__global__ void gat_logits(const float* __restrict__ h,
                           const float* __restrict__ a_src,
                           const float* __restrict__ a_dst,
                           float* __restrict__ e_src, float* __restrict__ e_dst,
                           int N, int H, int C) {
    int t = blockIdx.x * blockDim.x + threadIdx.x;
    if (t >= N * H) return;
    int n = t / H, hh = t - n * H;
    const float* hp = h + (size_t)n * H * C + (size_t)hh * C;
    const float* as = a_src + hh * C;
    const float* ad = a_dst + hh * C;
    float s = 0.f, d = 0.f;
    for (int c = 0; c < C; c += 4) {
        float4 hv = *(const float4*)(hp + c);
        float4 av = *(const float4*)(as + c);
        float4 dv = *(const float4*)(ad + c);
        s += hv.x * av.x + hv.y * av.y + hv.z * av.z + hv.w * av.w;
        d += hv.x * dv.x + hv.y * dv.y + hv.z * dv.z + hv.w * dv.w;
    }
    e_src[t] = s;
    e_dst[t] = d;
}

__device__ __forceinline__ float lrelu(float x, float slope) {
    return x > 0.f ? x : slope * x;
}

// Edge ids >= E are self loops (src = dst = e - E).
__device__ __forceinline__ void edge_sd(int e, int E, const int* __restrict__ s,
                                        const int* __restrict__ d, int& src, int& dst) {
    if (e < E) { src = s[e]; dst = d[e]; } else { src = e - E; dst = src; }
}

// ---------------------------------------------------------------------------
// Pass 1: segment max of leaky_relu(e_src[src]+e_dst[dst]) per (dst, head).
// ---------------------------------------------------------------------------
__global__ void gat_edge_max(const int* __restrict__ srcIdx, const int* __restrict__ dstIdx,
                             int E, int N, const float* __restrict__ e_src,
                             const float* __restrict__ e_dst,
                             unsigned* __restrict__ mOrd, int H) {
    int e = blockIdx.x * blockDim.x + threadIdx.x;
    if (e >= E + N) return;
    int s, d; edge_sd(e, E, srcIdx, dstIdx, s, d);
    for (int h = 0; h < H; ++h) {
        float a = lrelu(e_src[s * H + h] + e_dst[d * H + h], NEG_SLOPE_ATT);
        atomicMax(&mOrd[d * H + h], f2ord(a));
    }
}

// ---------------------------------------------------------------------------
// Pass 2: denom[dst,h] = sum exp(alpha - max).
// ---------------------------------------------------------------------------
__global__ void gat_edge_sum(const int* __restrict__ srcIdx, const int* __restrict__ dstIdx,
                             int E, int N, const float* __restrict__ e_src,
                             const float* __restrict__ e_dst,
                             const unsigned* __restrict__ mOrd,
                             float* __restrict__ denom, int H) {
    int e = blockIdx.x * blockDim.x + threadIdx.x;
    if (e >= E + N) return;
    int s, d; edge_sd(e, E, srcIdx, dstIdx, s, d);
    for (int h = 0; h < H; ++h) {
        float a = lrelu(e_src[s * H + h] + e_dst[d * H + h], NEG_SLOPE_ATT);
        float ex = __expf(a - ord2f(mOrd[d * H + h]));
        atomicAdd(&denom[d * H + h], ex);
    }
}

// ---------------------------------------------------------------------------
// Pass 3: out[dst,:] += coef * h[src,:]; one wave per edge, lane-parallel
// columns. Per-head coefficient computed once by lanes 0..H-1 and broadcast
// with __shfl (wave32 lane permute). h and out are L2-resident on MI455X.
// ---------------------------------------------------------------------------
__global__ __launch_bounds__(256)
void gat_edge_scatter(const int* __restrict__ srcIdx, const int* __restrict__ dstIdx,
                      int E, int N, const float* __restrict__ e_src,
                      const float* __restrict__ e_dst,
                      const unsigned* __restrict__ mOrd,
                      const float* __restrict__ denom,
                      const float* __restrict__ h, float* __restrict__ out,
                      int H, int C) {
    int wid  = (int)(((size_t)blockIdx.x * blockDim.x + threadIdx.x) >> 5);
    int lane = threadIdx.x & 31;
    if (wid >= E + N) return;
    int s, d; edge_sd(wid, E, srcIdx, dstIdx, s, d);
    const int F = H * C;

    __builtin_prefetch(&h[(size_t)s * F + lane], 0, 1);   // global_prefetch_b8

    // Lanes 0..H-1 each own one head's softmax coefficient.
    float coefLane = 0.f;
    if (lane < H) {
        float a = lrelu(e_src[s * H + lane] + e_dst[d * H + lane], NEG_SLOPE_ATT);
        coefLane = __expf(a - ord2f(mOrd[d * H + lane])) / denom[d * H + lane];
    }
    for (int col = lane; col < F; col += 32) {
        float coef = __shfl(coefLane, col / C, 32);       // head id is wave-uniform
        atomicAdd(&out[(size_t)d * F + col], h[(size_t)s * F + col] * coef);
    }
}

// ---------------------------------------------------------------------------
// x = (x + bias[col]), optional leaky-relu.
// ---------------------------------------------------------------------------
__global__ void gat_bias_act(float* __restrict__ x, const float* __restrict__ b,
                             size_t total, int F, float slope, int apply_act) {
    size_t i = (size_t)blockIdx.x * blockDim.x + threadIdx.x;
    if (i >= total) return;
    int c = (int)(i % F);
    float v = x[i] + b[c];
    if (apply_act) v = lrelu(v, slope);
    x[i] = v;
}

// ---------------------------------------------------------------------------
// Host-side orchestration.
// ---------------------------------------------------------------------------
extern "C" void kernel_launch(void* const* d_in, const int* in_sizes, int n_in,
                              void* d_out, int out_size, void* d_ws, size_t ws_size,
                              hipStream_t stream) {
    const float* x        = (const float*)d_in[0];
    const int*   ei       = (const int*)d_in[1];
    const float* W1       = (const float*)d_in[2];
    const float* att_src1 = (const float*)d_in[3];
    const float* att_dst1 = (const float*)d_in[4];
    const float* b1       = (const float*)d_in[5];
    const float* W2       = (const float*)d_in[6];
    const float* att_src2 = (const float*)d_in[7];
    const float* att_dst2 = (const float*)d_in[8];
    const float* b2       = (const float*)d_in[9];

    const int HC1  = in_sizes[5];            // H1*C1 = 128
    const int OUTC = in_sizes[9];            // 64
    const int F_IN = in_sizes[2] / HC1;      // 128
    const int N    = in_sizes[0] / F_IN;     // 100000
    const int E    = in_sizes[1] / 2;        // 1600000
    const int H1 = 4, C1 = HC1 / 4;          // GATConv(heads=4)
    const int H2 = 1, C2 = OUTC;             // GATConv(heads=1)
    const int Etot = E + N;                  // + self loops

    const int* srcIdx = ei;
    const int* dstIdx = ei + E;

    // Workspace layout (floats). h2 aliases h1 (dead after layer-1 scatter).
    float* ws  = (float*)d_ws;
    size_t off = 0;
    float* h1   = ws + off; off += (size_t)N * HC1;   // layer-1 projection
    float* act1 = ws + off; off += (size_t)N * HC1;   // layer-1 aggregated output
    float* es1  = ws + off; off += (size_t)N * H1;
    float* ed1  = ws + off; off += (size_t)N * H1;
    float* dn1  = ws + off; off += (size_t)N * H1;
    unsigned* m1 = (unsigned*)(ws + off); off += (size_t)N * H1;
    float* es2  = ws + off; off += (size_t)N;
    float* ed2  = ws + off; off += (size_t)N;
    float* dn2  = ws + off; off += (size_t)N;
    unsigned* m2 = (unsigned*)(ws + off); off += (size_t)N;
    float* h2   = h1;                                  // alias: N*OUTC <= N*HC1
    float* out2 = (float*)d_out;                       // [N, OUTC]

    const int TB = 256;
    const int fillBlocks = 2048;

    // --- init (d_out / ws are poisoned by harness) ---
    gat_fill_u32<<<fillBlocks, TB, 0, stream>>>((unsigned*)act1, 0u, (size_t)N * HC1);
    gat_fill_u32<<<fillBlocks, TB, 0, stream>>>((unsigned*)out2, 0u, (size_t)N * OUTC);
    gat_fill_u32<<<fillBlocks, TB, 0, stream>>>((unsigned*)dn1, 0u, (size_t)N * H1);
    gat_fill_u32<<<fillBlocks, TB, 0, stream>>>(m1, ORD_NEG_INF, (size_t)N * H1);
    gat_fill_u32<<<fillBlocks, TB, 0, stream>>>((unsigned*)dn2, 0u, (size_t)N);
    gat_fill_u32<<<fillBlocks, TB, 0, stream>>>(m2, ORD_NEG_INF, (size_t)N);

    const int rowTiles32 = (N + 31) / 32;
    const int edgeBlocks = (Etot + TB - 1) / TB;
    const int scatterBlocks = (int)(((size_t)Etot * 32 + TB - 1) / TB);

    // ================= Layer 1 =================
    gat_gemm_wmma<<<rowTiles32, 32 * (HC1 / 16), 32 * (F_IN + 4) * sizeof(float), stream>>>(
        x, W1, h1, N, F_IN, HC1);
    gat_logits<<<(N * H1 + TB - 1) / TB, TB, 0, stream>>>(
        h1, att_src1, att_dst1, es1, ed1, N, H1, C1);
    gat_edge_max<<<edgeBlocks, TB, 0, stream>>>(srcIdx, dstIdx, E, N, es1, ed1, m1, H1);
    gat_edge_sum<<<edgeBlocks, TB, 0, stream>>>(srcIdx, dstIdx, E, N, es1, ed1, m1, dn1, H1);
    gat_edge_scatter<<<scatterBlocks, TB, 0, stream>>>(
        srcIdx, dstIdx, E, N, es1, ed1, m1, dn1, h1, act1, H1, C1);
    gat_bias_act<<<(int)(((size_t)N * HC1 + TB - 1) / TB), TB, 0, stream>>>(
        act1, b1, (size_t)N * HC1, HC1, NEG_SLOPE_ACT, /*apply_act=*/1);

    // ================= Layer 2 =================
    gat_gemm_wmma<<<rowTiles32, 32 * (OUTC / 16), 32 * (HC1 + 4) * sizeof(float), stream>>>(
        act1, W2, h2, N, HC1, OUTC);
    gat_logits<<<(N * H2 + TB - 1) / TB, TB, 0, stream>>>(
        h2, att_src2, att_dst2, es2, ed2, N, H2, C2);
    gat_edge_max<<<edgeBlocks, TB, 0, stream>>>(srcIdx, dstIdx, E, N, es2, ed2, m2, H2);
    gat_edge_sum<<<edgeBlocks, TB, 0, stream>>>(srcIdx, dstIdx, E, N, es2, ed2, m2, dn2, H2);
    gat_edge_scatter<<<scatterBlocks, TB, 0, stream>>>(
        srcIdx, dstIdx, E, N, es2, ed2, m2, dn2, h2, out2, H2, C2);
    gat_bias_act<<<(int)(((size_t)N * OUTC + TB - 1) / TB), TB, 0, stream>>>(
        out2, b2, (size_t)N * OUTC, OUTC, 0.f, /*apply_act=*/0);
}